// OrientationLossV1_60842506715476
// MI455X (gfx1250) — compile-verified
//
#include <hip/hip_runtime.h>
#include <hip/hip_bf16.h>

// ---------------------------------------------------------------------------
// OrientationLoss for MI455X (gfx1250), compile-only target.
//
// loss = (1/(Ns^2*Nt)) * sum_t [ ||S_t||^2 - 2 * sum_g ||S_tg||^2 ]
//   u_ti = (t_t - s_i)/n_ti,  S_t = sum_i u_ti,  S_tg = sum_{i in group g} u_ti
// Heavy lifting = two f32 WMMA GEMMs:
//   C = trgt @ src^T   (dots; norms via ||t||^2 - 2C + ||s||^2)
//   Y = W @ src        (W[t,i] = 1/n_ti; gives y_t = sum_i w_ti s_i)
// ---------------------------------------------------------------------------

typedef __attribute__((ext_vector_type(2))) float v2f;
typedef __attribute__((ext_vector_type(8))) float v8f;

#define NS 384
#define NT 384
#define DD 256
#define NG 96   // groups of 4

// ---------------- row squared norms (one wave per row) ----------------------
__global__ void row_norm2_kernel(const float* __restrict__ X, float* __restrict__ out, int D) {
  int lane = threadIdx.x & 31;
  int row  = (blockIdx.x * blockDim.x + threadIdx.x) >> 5;
  const float* x = X + (size_t)row * D;
  float s = 0.f;
  for (int d = lane; d < D; d += 32) { float v = x[d]; s += v * v; }
#pragma unroll
  for (int m = 16; m >= 1; m >>= 1) s += __shfl_xor(s, m, 32);
  if (lane == 0) out[row] = s;
}

// ---------------- WMMA GEMM, C[m,n] = sum_k A[m,k]*B[n,k] (A,B row-major, "NT")
// One wave per 16x16 tile of C. f32 16x16x4 WMMA.
// A-layout per ISA: lanes 0-15 -> M=lane, VGPR{0,1}=K{0,1}; lanes 16-31 -> K{2,3}.
__global__ void gemm_nt_wmma(const float* __restrict__ A, const float* __restrict__ B,
                             float* __restrict__ C, int M, int N, int K) {
  int wave = (int)((blockIdx.x * blockDim.x + threadIdx.x) >> 5);
  int lane = threadIdx.x & 31;
  int tilesN = N >> 4;
  int tm = (wave / tilesN) << 4;
  int tn = (wave % tilesN) << 4;
  if (tm >= M) return;                       // wave-uniform
  int r  = lane & 15;
  int kk = (lane >> 4) << 1;                 // 0 or 2
  const float* arow = A + (size_t)(tm + r) * K + kk;
  const float* brow = B + (size_t)(tn + r) * K + kk;
  v8f acc = {};
#pragma unroll 4
  for (int k = 0; k < K; k += 4) {
    v2f a = *(const v2f*)(arow + k);
    v2f b = *(const v2f*)(brow + k);
    acc = __builtin_amdgcn_wmma_f32_16x16x4_f32(false, a, false, b, (short)0, acc,
                                                false, false);
  }
  int cm = tm + ((lane >> 4) << 3);          // C/D: vgpr j -> row j (+8 for hi half)
  int cn = tn + r;
#pragma unroll
  for (int j = 0; j < 8; ++j)
    C[(size_t)(cm + j) * N + cn] = acc[j];
}

// ---------------- WMMA GEMM, C[m,n] = sum_k A[m,k]*B[k,n] ("NN") ------------
__global__ void gemm_nn_wmma(const float* __restrict__ A, const float* __restrict__ B,
                             float* __restrict__ C, int M, int N, int K) {
  int wave = (int)((blockIdx.x * blockDim.x + threadIdx.x) >> 5);
  int lane = threadIdx.x & 31;
  int tilesN = N >> 4;
  int tm = (wave / tilesN) << 4;
  int tn = (wave % tilesN) << 4;
  if (tm >= M) return;                       // wave-uniform
  int r  = lane & 15;
  int kk = (lane >> 4) << 1;
  const float* arow = A + (size_t)(tm + r) * K + kk;
  int bcol = tn + r;
  v8f acc = {};
#pragma unroll 4
  for (int k = 0; k < K; k += 4) {
    v2f a = *(const v2f*)(arow + k);
    v2f b;
    b.x = B[(size_t)(k + kk) * N + bcol];
    b.y = B[(size_t)(k + kk + 1) * N + bcol];
    acc = __builtin_amdgcn_wmma_f32_16x16x4_f32(false, a, false, b, (short)0, acc,
                                                false, false);
  }
  int cm = tm + ((lane >> 4) << 3);
  int cn = tn + r;
#pragma unroll
  for (int j = 0; j < 8; ++j)
    C[(size_t)(cm + j) * N + cn] = acc[j];
}

// ---------------- W[t,i] = 1/n_ti, n^2 = ||t||^2 + ||s||^2 - 2*dot ----------
__global__ void make_w_kernel(const float* __restrict__ C, const float* __restrict__ tn2,
                              const float* __restrict__ sn2, float* __restrict__ W, int N) {
  int idx = blockIdx.x * blockDim.x + threadIdx.x;
  int t = idx / N, i = idx - t * N;
  float n2 = tn2[t] + sn2[i] - 2.f * C[idx];
  n2 = fmaxf(n2, 1e-8f);                     // matches max(n_i*n_j, eps) regime
  W[idx] = rsqrtf(n2);
}

// ---------------- per-group 4x4 src Grams (block diagonal only) -------------
__global__ void group_gram_kernel(const float* __restrict__ src, float* __restrict__ G4, int D) {
  int idx = blockIdx.x * blockDim.x + threadIdx.x;   // g*16 + i*4 + j
  if (idx >= NG * 16) return;
  int g = idx >> 4;
  int i = (idx >> 2) & 3;
  int j = idx & 3;
  const float* a = src + (size_t)(g * 4 + i) * D;
  const float* b = src + (size_t)(g * 4 + j) * D;
  float s = 0.f;
  for (int d = 0; d < D; ++d) s += a[d] * b[d];
  G4[idx] = s;
}

// ---------------- per-target reduction --------------------------------------
// loss_t = [a^2*||t||^2 - 2 a p + ||y||^2]  -  2 * sum_g [a_g^2*||t||^2 - 2 a_g p_g + gg_g]
//   a = sum_i w,  p = sum_i w*c  (= y.t),  y from GEMM2,  gg_g from 4x4 Grams.
__global__ void per_t_loss_kernel(const float* __restrict__ W, const float* __restrict__ C,
                                  const float* __restrict__ Y, const float* __restrict__ tn2v,
                                  const float* __restrict__ G4, float* __restrict__ tloss,
                                  int Nsrc, int D) {
  int t = blockIdx.x;
  int tid = threadIdx.x;                     // blockDim = 128
  __shared__ float s_grp[128], s_a[128], s_p[128], s_q[128];

  float tn2 = tn2v[t];
  float grp = 0.f, asum = 0.f, psum = 0.f;
  if (tid < NG) {
    int g = tid;
    float w[4], c[4];
#pragma unroll
    for (int k = 0; k < 4; ++k) {
      w[k] = W[(size_t)t * Nsrc + 4 * g + k];
      c[k] = C[(size_t)t * Nsrc + 4 * g + k];
    }
    float ag = 0.f, pg = 0.f, gg = 0.f;
#pragma unroll
    for (int a = 0; a < 4; ++a) {
      ag += w[a];
      pg += w[a] * c[a];
#pragma unroll
      for (int b = 0; b < 4; ++b) gg += w[a] * w[b] * G4[g * 16 + a * 4 + b];
    }
    grp  = ag * ag * tn2 - 2.f * ag * pg + gg;
    asum = ag;
    psum = pg;
  }
  float q = 0.f;
  for (int d = tid; d < D; d += blockDim.x) { float y = Y[(size_t)t * D + d]; q += y * y; }

  s_grp[tid] = grp; s_a[tid] = asum; s_p[tid] = psum; s_q[tid] = q;
  __syncthreads();
  for (int off = 64; off >= 1; off >>= 1) {
    if (tid < off) {
      s_grp[tid] += s_grp[tid + off];
      s_a[tid]   += s_a[tid + off];
      s_p[tid]   += s_p[tid + off];
      s_q[tid]   += s_q[tid + off];
    }
    __syncthreads();
  }
  if (tid == 0) {
    float a = s_a[0], p = s_p[0];
    float total = a * a * tn2 - 2.f * a * p + s_q[0];
    tloss[t] = total - 2.f * s_grp[0];
  }
}

// ---------------- final scalar ----------------------------------------------
__global__ void final_reduce_kernel(const float* __restrict__ tloss, float* __restrict__ out,
                                    int Nt, float scale) {
  __shared__ float red[256];
  int tid = threadIdx.x;
  float s = 0.f;
  for (int i = tid; i < Nt; i += blockDim.x) s += tloss[i];
  red[tid] = s;
  __syncthreads();
  for (int off = 128; off >= 1; off >>= 1) {
    if (tid < off) red[tid] += red[tid + off];
    __syncthreads();
  }
  if (tid == 0) out[0] = red[0] * scale;
}

// ---------------------------------------------------------------------------
extern "C" void kernel_launch(void* const* d_in, const int* in_sizes, int n_in,
                              void* d_out, int out_size, void* d_ws, size_t ws_size,
                              hipStream_t stream) {
  (void)in_sizes; (void)n_in; (void)out_size; (void)ws_size;
  const float* src  = (const float*)d_in[0];   // [384, 256]
  const float* trgt = (const float*)d_in[1];   // [384, 256]
  float* out = (float*)d_out;                  // scalar f32

  // workspace layout (floats)
  float* ws    = (float*)d_ws;
  float* Cmat  = ws;                            // 384*384
  float* Wmat  = Cmat + NT * NS;                // 384*384
  float* Ymat  = Wmat + NT * NS;                // 384*256
  float* tn2   = Ymat + NT * DD;                // 384
  float* sn2   = tn2 + NT;                      // 384
  float* G4    = sn2 + NS;                      // 96*16
  float* tloss = G4 + NG * 16;                  // 384

  // 1) row squared norms (8 rows per 256-thread block)
  row_norm2_kernel<<<NT / 8, 256, 0, stream>>>(trgt, tn2, DD);
  row_norm2_kernel<<<NS / 8, 256, 0, stream>>>(src,  sn2, DD);

  // 2) C = trgt @ src^T  : 24x24 tiles = 576 waves = 144 blocks of 128
  gemm_nt_wmma<<<(NT / 16) * (NS / 16) / 4, 128, 0, stream>>>(trgt, src, Cmat, NT, NS, DD);

  // 3) W = 1/n
  make_w_kernel<<<(NT * NS) / 256, 256, 0, stream>>>(Cmat, tn2, sn2, Wmat, NS);

  // 4) per-group 4x4 src Grams
  group_gram_kernel<<<(NG * 16 + 255) / 256, 256, 0, stream>>>(src, G4, DD);

  // 5) Y = W @ src : 24x16 tiles = 384 waves = 96 blocks of 128
  gemm_nn_wmma<<<(NT / 16) * (DD / 16) / 4, 128, 0, stream>>>(Wmat, src, Ymat, NT, DD, NS);

  // 6) per-t loss
  per_t_loss_kernel<<<NT, 128, 0, stream>>>(Wmat, Cmat, Ymat, tn2, G4, tloss, NS, DD);

  // 7) final scalar
  const float scale = 1.0f / ((float)NS * (float)NS * (float)NT);
  final_reduce_kernel<<<1, 256, 0, stream>>>(tloss, out, NT, scale);
}